// SVDLayer_49271864820310
// MI455X (gfx1250) — compile-verified
//
#include <hip/hip_runtime.h>
#include <hip/hip_bf16.h>

typedef __attribute__((ext_vector_type(2))) float v2f;
typedef __attribute__((ext_vector_type(8))) float v8f;

#define BATCHN 8192
#define DD 64
#define PP 65          // LDS pitch padded: stride 65 words -> conflict-free across 64 banks
#define NSWEEP 10

__launch_bounds__(256, 1)
__global__ void svd64_jacobi_wmma(const float* __restrict__ A,
                                  float* __restrict__ outU,
                                  float* __restrict__ outS,
                                  float* __restrict__ outV) {
  __shared__ float As[DD * PP];   // input matrix, row-major padded
  __shared__ float Gs[DD * PP];   // Gram matrix A^T A
  __shared__ float Vs[DD * PP];   // accumulated right singular vectors
  __shared__ float cA[32], sA[32];
  __shared__ int   pA[32], qA[32];
  __shared__ float valA[DD];
  __shared__ int   idxA[DD];
  __shared__ float rinvA[DD];

  const int tid  = threadIdx.x;
  const int lane = tid & 31;
  const int wave = tid >> 5;
  const long b   = blockIdx.x;
  const float* Ab = A + b * (long)(DD * DD);

  // ---- load A into LDS; init V = I ----
  for (int e = tid; e < DD * DD; e += 256) {
    int r = e >> 6, c = e & 63;
    As[r * PP + c] = Ab[e];
    Vs[r * PP + c] = (r == c) ? 1.0f : 0.0f;
  }
  __syncthreads();

  // ---- G = A^T * A via V_WMMA_F32_16X16X4_F32 (2 tiles per wave) ----
  const int m  = lane & 15;            // M (and N) index within tile
  const int kh = (lane >> 4) << 1;     // K sub-offset: 0 or 2

  for (int t = 0; t < 2; ++t) {
    const int tile = wave * 2 + t;
    const int I = tile >> 2, J = tile & 3;
    v8f acc = {};
#pragma unroll
    for (int kk = 0; kk < DD; kk += 4) {
      v2f a, bm;
      // A-operand = (A^T) tile: Amat[m][k] = A[kk+k][I*16+m]
      a.x  = As[(kk + kh    ) * PP + I * 16 + m];
      a.y  = As[(kk + kh + 1) * PP + I * 16 + m];
      // B-operand = A tile: Bmat[k][n] = A[kk+k][J*16+n], n == m == lane&15
      bm.x = As[(kk + kh    ) * PP + J * 16 + m];
      bm.y = As[(kk + kh + 1) * PP + J * 16 + m];
      acc = __builtin_amdgcn_wmma_f32_16x16x4_f32(false, a, false, bm,
                                                  (short)0, acc, false, false);
    }
#pragma unroll
    for (int v = 0; v < 8; ++v)
      Gs[(I * 16 + v + ((lane >> 4) << 3)) * PP + J * 16 + m] = acc[v];
  }
  __syncthreads();

  // ---- cyclic parallel Jacobi (round-robin tournament pairing) ----
  for (int sweep = 0; sweep < NSWEEP; ++sweep) {
    for (int r = 0; r < DD - 1; ++r) {
      if (tid < 32) {
        int p, q;
        if (tid == 0) { p = DD - 1; q = r; }
        else {
          p = (r + tid) % (DD - 1);
          q = (r - tid + (DD - 1)) % (DD - 1);
        }
        float app = Gs[p * PP + p];
        float aqq = Gs[q * PP + q];
        float apq = Gs[p * PP + q];
        float c = 1.0f, s = 0.0f;
        if (fabsf(apq) > 1e-30f) {
          float tau = (aqq - app) / (2.0f * apq);
          float tt  = (tau >= 0.0f ? 1.0f : -1.0f) /
                      (fabsf(tau) + sqrtf(1.0f + tau * tau));
          c = rsqrtf(1.0f + tt * tt);
          s = tt * c;
        }
        pA[tid] = p; qA[tid] = q; cA[tid] = c; sA[tid] = s;
      }
      __syncthreads();

      const int   pr = pA[tid >> 3], qr = qA[tid >> 3];
      const float c  = cA[tid >> 3], s  = sA[tid >> 3];
      const int   i0 = (tid & 7) << 3;

      // right-multiply: columns p,q of G and V   (G <- G*J, V <- V*J)
#pragma unroll
      for (int ii = 0; ii < 8; ++ii) {
        int i = i0 + ii;
        float gp = Gs[i * PP + pr], gq = Gs[i * PP + qr];
        Gs[i * PP + pr] = c * gp - s * gq;
        Gs[i * PP + qr] = s * gp + c * gq;
        float vp = Vs[i * PP + pr], vq = Vs[i * PP + qr];
        Vs[i * PP + pr] = c * vp - s * vq;
        Vs[i * PP + qr] = s * vp + c * vq;
      }
      __syncthreads();

      // left-multiply: rows p,q of G   (G <- J^T * G)
#pragma unroll
      for (int ii = 0; ii < 8; ++ii) {
        int i = i0 + ii;
        float gp = Gs[pr * PP + i], gq = Gs[qr * PP + i];
        Gs[pr * PP + i] = c * gp - s * gq;
        Gs[qr * PP + i] = s * gp + c * gq;
      }
      __syncthreads();
    }
  }

  // ---- singular values + bitonic sort descending (value,index) ----
  if (tid < DD) {
    valA[tid] = sqrtf(fmaxf(Gs[tid * PP + tid], 0.0f));
    idxA[tid] = tid;
  }
  __syncthreads();
  for (int k = 2; k <= DD; k <<= 1) {
    for (int j = k >> 1; j > 0; j >>= 1) {
      if (tid < DD) {
        int l = tid ^ j;
        if (l > tid) {
          float v0 = valA[tid], v1 = valA[l];
          bool descBlock = ((tid & k) == 0);
          if ((descBlock && v0 < v1) || (!descBlock && v0 > v1)) {
            valA[tid] = v1; valA[l] = v0;
            int t0 = idxA[tid]; idxA[tid] = idxA[l]; idxA[l] = t0;
          }
        }
      }
      __syncthreads();
    }
  }
  if (tid < DD) {
    float sv = valA[tid];
    rinvA[tid] = (sv > 1e-20f) ? (1.0f / sv) : 0.0f;
  }
  __syncthreads();

  // ---- U = A * V(:,perm) / s  via WMMA; write outputs ----
  float* Ub = outU + b * (long)(DD * DD);
  float* Sb = outS + b * (long)(DD * DD);
  float* Vb = outV + b * (long)(DD * DD);

  for (int t = 0; t < 2; ++t) {
    const int tile = wave * 2 + t;
    const int I = tile >> 2, J = tile & 3;
    const int   coln = idxA[J * 16 + m];   // permuted V column (sorted order)
    const float rin  = rinvA[J * 16 + m];
    v8f acc = {};
#pragma unroll
    for (int kk = 0; kk < DD; kk += 4) {
      v2f a, bm;
      // Amat[m][k] = A[I*16+m][kk+k]  (transposed read -> padded pitch avoids conflicts)
      a.x  = As[(I * 16 + m) * PP + kk + kh];
      a.y  = As[(I * 16 + m) * PP + kk + kh + 1];
      // Bmat[k][n] = V[kk+k][idx[J*16+n]]
      bm.x = Vs[(kk + kh    ) * PP + coln];
      bm.y = Vs[(kk + kh + 1) * PP + coln];
      acc = __builtin_amdgcn_wmma_f32_16x16x4_f32(false, a, false, bm,
                                                  (short)0, acc, false, false);
    }
#pragma unroll
    for (int v = 0; v < 8; ++v)
      Ub[(I * 16 + v + ((lane >> 4) << 3)) * DD + J * 16 + m] = acc[v] * rin;
  }

  for (int e = tid; e < DD * DD; e += 256) {
    int r = e >> 6, c = e & 63;
    Sb[e] = (r == c) ? valA[r] : 0.0f;
    Vb[e] = Vs[r * PP + idxA[c]];
  }
}

extern "C" void kernel_launch(void* const* d_in, const int* in_sizes, int n_in,
                              void* d_out, int out_size, void* d_ws, size_t ws_size,
                              hipStream_t stream) {
  (void)in_sizes; (void)n_in; (void)d_ws; (void)ws_size; (void)out_size;
  const float* A = (const float*)d_in[0];
  float* out = (float*)d_out;
  const long N = (long)BATCHN * DD * DD;
  svd64_jacobi_wmma<<<BATCHN, 256, 0, stream>>>(A, out, out + N, out + 2 * N);
}